// FastRGCN_71932112273608
// MI455X (gfx1250) — compile-verified
//
#include <hip/hip_runtime.h>
#include <hip/hip_bf16.h>

typedef __attribute__((ext_vector_type(2))) float v2f;
typedef __attribute__((ext_vector_type(8))) float v8f;

#define N_NODES 40000
#define N_EDGES 400000
#define IN_CH   64
#define HID     16
#define N_REL   90
#define N_BASES 30
#define N_CLASSES 8

// ---------------- utility kernels ----------------

__global__ void zero_u32(unsigned* __restrict__ p, int n) {
    int i = blockIdx.x * blockDim.x + threadIdx.x;
    if (i < n) p[i] = 0u;
}

// counts per (dst, relation) segment
__global__ void count_edges(const int* __restrict__ dst, const int* __restrict__ et,
                            unsigned* __restrict__ cnt, int E) {
    int e = blockIdx.x * blockDim.x + threadIdx.x;
    if (e < E) atomicAdd(&cnt[dst[e] * N_REL + et[e]], 1u);
}

// W1[r][i][o] = sum_b comp1[r,b] * basis1[b,i,o]   (90 x 64 x 16)
__global__ void make_w1(const float* __restrict__ comp, const float* __restrict__ basis,
                        float* __restrict__ W) {
    int idx = blockIdx.x * blockDim.x + threadIdx.x;
    if (idx >= N_REL * IN_CH * HID) return;
    int r = idx / (IN_CH * HID);
    int rem = idx - r * (IN_CH * HID);   // i*HID + o
    float acc = 0.f;
    #pragma unroll 5
    for (int b = 0; b < N_BASES; ++b)
        acc += comp[r * N_BASES + b] * basis[b * (IN_CH * HID) + rem];
    W[idx] = acc;
}

// W2[r][i][o] = sum_b comp2[r,b] * basis2[b,i,o]   (90 x 16 x 8)
__global__ void make_w2(const float* __restrict__ comp, const float* __restrict__ basis,
                        float* __restrict__ W) {
    int idx = blockIdx.x * blockDim.x + threadIdx.x;
    if (idx >= N_REL * HID * N_CLASSES) return;
    int r = idx / (HID * N_CLASSES);
    int rem = idx - r * (HID * N_CLASSES);
    float acc = 0.f;
    #pragma unroll 5
    for (int b = 0; b < N_BASES; ++b)
        acc += comp[r * N_BASES + b] * basis[b * (HID * N_CLASSES) + rem];
    W[idx] = acc;
}

// pad root2 [16x8] -> root2p [16x16] with zero columns 8..15
__global__ void pad_root2(const float* __restrict__ root2, float* __restrict__ root2p) {
    int idx = threadIdx.x;           // one block of 256
    if (idx >= HID * 16) return;
    int k = idx >> 4, n = idx & 15;
    root2p[idx] = (n < N_CLASSES) ? root2[k * N_CLASSES + n] : 0.0f;
}

// ---------------- WMMA root transform 1: h = x @ root1 + bias1 ----------------
// One wave32 per 16-node tile. K=64 as 16 chained V_WMMA_F32_16X16X4_F32.
// A (16x4 f32): v0: lanes0-15 K=k, lanes16-31 K=k+2; v1: K=k+1 / K=k+3.
// B (4x16 f32): mirrored, N striped across lanes.
// C/D (16x16 f32): VGPR j -> (M = j + 8*(lane>=16), N = lane&15).
__global__ void __launch_bounds__(256)
root_gemm1(const float* __restrict__ x, const float* __restrict__ root,
           const float* __restrict__ bias, float* __restrict__ h) {
    const int ntiles = N_NODES / 16;  // 2500
    int wave = (int)((blockIdx.x * blockDim.x + threadIdx.x) >> 5);
    int lane = (int)(threadIdx.x & 31);
    if (wave >= ntiles) return;

    int l15   = lane & 15;          // A row (M) / B col (N)
    int khalf = (lane >> 4) * 2;    // 0 for lanes 0-15, 2 for lanes 16-31
    int m0    = wave * 16;

    const float* xrow = x + (size_t)(m0 + l15) * IN_CH;
    v8f acc = {};
    #pragma unroll
    for (int k0 = 0; k0 < IN_CH; k0 += 4) {
        v2f a = *(const v2f*)(xrow + k0 + khalf);                 // x[m][k], x[m][k+1]
        v2f b;
        b.x = root[(k0 + khalf)     * HID + l15];                 // root[k][n]
        b.y = root[(k0 + khalf + 1) * HID + l15];
        acc = __builtin_amdgcn_wmma_f32_16x16x4_f32(
                  false, a, false, b, (short)0, acc, false, false);
    }
    float bz = bias[l15];
    int mbase = m0 + ((lane >> 4) * 8);
    #pragma unroll
    for (int j = 0; j < 8; ++j)
        h[(size_t)(mbase + j) * HID + l15] = acc[j] + bz;
}

// ---------------- WMMA root transform 2: out = h @ root2p + bias2 ----------------
// Same tiling, K=16 (4 WMMA steps), only columns 0..7 are stored.
__global__ void __launch_bounds__(256)
root_gemm2(const float* __restrict__ h, const float* __restrict__ root2p,
           const float* __restrict__ bias, float* __restrict__ out) {
    const int ntiles = N_NODES / 16;  // 2500
    int wave = (int)((blockIdx.x * blockDim.x + threadIdx.x) >> 5);
    int lane = (int)(threadIdx.x & 31);
    if (wave >= ntiles) return;

    int l15   = lane & 15;
    int khalf = (lane >> 4) * 2;
    int m0    = wave * 16;

    const float* hrow = h + (size_t)(m0 + l15) * HID;
    v8f acc = {};
    #pragma unroll
    for (int k0 = 0; k0 < HID; k0 += 4) {
        v2f a = *(const v2f*)(hrow + k0 + khalf);
        v2f b;
        b.x = root2p[(k0 + khalf)     * 16 + l15];
        b.y = root2p[(k0 + khalf + 1) * 16 + l15];
        acc = __builtin_amdgcn_wmma_f32_16x16x4_f32(
                  false, a, false, b, (short)0, acc, false, false);
    }
    if (l15 < N_CLASSES) {
        float bz = bias[l15];
        int mbase = m0 + ((lane >> 4) * 8);
        #pragma unroll
        for (int j = 0; j < 8; ++j)
            out[(size_t)(mbase + j) * N_CLASSES + l15] = acc[j] + bz;
    }
}

// ---------------- edge messages, layer 1 ----------------
// 16 threads per edge; source rows staged through LDS (1 cooperative 16B load
// per thread instead of 16x redundant 256B row reads).
__global__ void __launch_bounds__(256)
edge_msg1(const int* __restrict__ src, const int* __restrict__ dst,
          const int* __restrict__ et, const float* __restrict__ x,
          const float* __restrict__ W1, const unsigned* __restrict__ cnt,
          float* __restrict__ h, int E) {
    __shared__ __align__(16) float xs[16 * IN_CH];   // 16 edges x 64 floats = 4 KB
    int tid    = (int)threadIdx.x;
    int eLocal = tid >> 4;          // 0..15
    int o      = tid & 15;          // output channel
    int e      = (int)blockIdx.x * 16 + eLocal;
    bool valid = e < E;
    int s = 0, d = 0, r = 0;
    if (valid) { s = src[e]; d = dst[e]; r = et[e]; }

    // stage: lane o loads x[src][4o .. 4o+3]
    if (valid) {
        float4 xv = *(const float4*)(x + (size_t)s * IN_CH + o * 4);
        *(float4*)(xs + eLocal * IN_CH + o * 4) = xv;
    }
    __syncthreads();
    if (!valid) return;

    float c    = (float)cnt[d * N_REL + r];
    float norm = 1.0f / fmaxf(c, 1.0f);
    const float* xr = xs + eLocal * IN_CH;
    const float* w  = W1 + (size_t)r * (IN_CH * HID) + o;
    float acc = 0.f;
    #pragma unroll
    for (int k = 0; k < IN_CH; ++k)
        acc = fmaf(xr[k], w[(size_t)k * HID], acc);
    atomicAdd(&h[(size_t)d * HID + o], acc * norm);
}

__global__ void relu_inplace(float* __restrict__ h, int n) {
    int i = blockIdx.x * blockDim.x + threadIdx.x;
    if (i < n) h[i] = fmaxf(h[i], 0.0f);
}

// ---------------- edge messages, layer 2 ----------------
// 8 threads per edge; hidden rows staged through LDS.
__global__ void __launch_bounds__(256)
edge_msg2(const int* __restrict__ src, const int* __restrict__ dst,
          const int* __restrict__ et, const float* __restrict__ h,
          const float* __restrict__ W2, const unsigned* __restrict__ cnt,
          float* __restrict__ out, int E) {
    __shared__ __align__(8) float hs[32 * HID];      // 32 edges x 16 floats = 2 KB
    int tid    = (int)threadIdx.x;
    int eLocal = tid >> 3;          // 0..31
    int c      = tid & 7;           // class
    int e      = (int)blockIdx.x * 32 + eLocal;
    bool valid = e < E;
    int s = 0, d = 0, r = 0;
    if (valid) { s = src[e]; d = dst[e]; r = et[e]; }

    if (valid) {
        float2 hv = *(const float2*)(h + (size_t)s * HID + c * 2);
        *(float2*)(hs + eLocal * HID + c * 2) = hv;
    }
    __syncthreads();
    if (!valid) return;

    float cn   = (float)cnt[d * N_REL + r];
    float norm = 1.0f / fmaxf(cn, 1.0f);
    const float* hr = hs + eLocal * HID;
    const float* w  = W2 + (size_t)r * (HID * N_CLASSES) + c;
    float acc = 0.f;
    #pragma unroll
    for (int k = 0; k < HID; ++k)
        acc = fmaf(hr[k], w[k * N_CLASSES], acc);
    atomicAdd(&out[(size_t)d * N_CLASSES + c], acc * norm);
}

// in-place log_softmax over the 8 classes
__global__ void log_softmax8(float* __restrict__ out) {
    int n = blockIdx.x * blockDim.x + threadIdx.x;
    if (n >= N_NODES) return;
    float v[N_CLASSES];
    float m = -3.402823466e+38f;
    #pragma unroll
    for (int c = 0; c < N_CLASSES; ++c) {
        v[c] = out[(size_t)n * N_CLASSES + c];
        m = fmaxf(m, v[c]);
    }
    float sum = 0.f;
    #pragma unroll
    for (int c = 0; c < N_CLASSES; ++c) sum += __expf(v[c] - m);
    float lse = m + __logf(sum);
    #pragma unroll
    for (int c = 0; c < N_CLASSES; ++c)
        out[(size_t)n * N_CLASSES + c] = v[c] - lse;
}

// ---------------- launcher ----------------
extern "C" void kernel_launch(void* const* d_in, const int* in_sizes, int n_in,
                              void* d_out, int out_size, void* d_ws, size_t ws_size,
                              hipStream_t stream) {
    const float* x      = (const float*)d_in[0];
    const int*   eidx   = (const int*)  d_in[1];
    const int*   etype  = (const int*)  d_in[2];
    const float* comp1  = (const float*)d_in[3];
    const float* basis1 = (const float*)d_in[4];
    const float* root1  = (const float*)d_in[5];
    const float* bias1  = (const float*)d_in[6];
    const float* comp2  = (const float*)d_in[7];
    const float* basis2 = (const float*)d_in[8];
    const float* root2  = (const float*)d_in[9];
    const float* bias2  = (const float*)d_in[10];
    float* out = (float*)d_out;

    const int* src = eidx;
    const int* dst = eidx + N_EDGES;

    // workspace layout (~17.4 MB)
    char* ws = (char*)d_ws;
    unsigned* cnt = (unsigned*)ws;                               // 3.6M u32 = 14.4 MB
    float* W1     = (float*)(ws + (size_t)N_NODES * N_REL * 4);  // 90*64*16 f32
    float* W2     = W1 + N_REL * IN_CH * HID;                    // 90*16*8 f32
    float* h      = W2 + N_REL * HID * N_CLASSES;                // 40000*16 f32
    float* root2p = h + (size_t)N_NODES * HID;                   // 16*16 f32 (padded)

    const int B = 256;
    // 1) zero (dst,rel) counters
    {
        int n = N_NODES * N_REL;
        zero_u32<<<(n + B - 1) / B, B, 0, stream>>>(cnt, n);
    }
    // 2) count edges per segment
    count_edges<<<(N_EDGES + B - 1) / B, B, 0, stream>>>(dst, etype, cnt, N_EDGES);
    // 3,4) fold bases into per-relation weights; pad root2
    {
        int n1 = N_REL * IN_CH * HID;
        make_w1<<<(n1 + B - 1) / B, B, 0, stream>>>(comp1, basis1, W1);
        int n2 = N_REL * HID * N_CLASSES;
        make_w2<<<(n2 + B - 1) / B, B, 0, stream>>>(comp2, basis2, W2);
        pad_root2<<<1, B, 0, stream>>>(root2, root2p);
    }
    // 5) h = x @ root1 + bias1  (WMMA, one wave per 16-node tile)
    {
        int ntiles = N_NODES / 16;                  // 2500
        int wavesPerBlock = B / 32;                 // 8
        int grid = (ntiles + wavesPerBlock - 1) / wavesPerBlock;
        root_gemm1<<<grid, B, 0, stream>>>(x, root1, bias1, h);
    }
    // 6) scatter layer-1 edge messages (16 edges per block, LDS-staged rows)
    edge_msg1<<<(N_EDGES + 15) / 16, B, 0, stream>>>(src, dst, etype, x, W1, cnt, h, N_EDGES);
    // 7) ReLU
    {
        int n = N_NODES * HID;
        relu_inplace<<<(n + B - 1) / B, B, 0, stream>>>(h, n);
    }
    // 8) logits = h @ root2 + bias2  (WMMA with zero-padded B)
    {
        int ntiles = N_NODES / 16;
        int wavesPerBlock = B / 32;
        int grid = (ntiles + wavesPerBlock - 1) / wavesPerBlock;
        root_gemm2<<<grid, B, 0, stream>>>(h, root2p, bias2, out);
    }
    // 9) scatter layer-2 edge messages (32 edges per block, LDS-staged rows)
    edge_msg2<<<(N_EDGES + 31) / 32, B, 0, stream>>>(src, dst, etype, h, W2, cnt, out, N_EDGES);
    // 10) in-place log_softmax
    log_softmax8<<<(N_NODES + B - 1) / B, B, 0, stream>>>(out);
}